// SiameseLoss_15839839388182
// MI455X (gfx1250) — compile-verified
//
#include <hip/hip_runtime.h>
#include <hip/hip_bf16.h>
#include <stdint.h>

// ---------------------------------------------------------------------------
// SiameseLoss on MI455X (gfx1250): memory-bound streaming reduction.
// 256 MiB of x1/x2 traffic @ 23.3 TB/s => ~11.5us floor; compute is trivial.
// CDNA5 path used: Tensor Data Mover (TENSOR_LOAD_TO_LDS) double-buffered
// global->LDS pipeline, S_WAIT_TENSORCNT, wave32 shuffle reductions.
// ---------------------------------------------------------------------------

typedef __attribute__((ext_vector_type(4))) unsigned int uint32x4;
typedef __attribute__((ext_vector_type(8))) int          int32x8;
typedef __attribute__((ext_vector_type(4))) int          int32x4;

static constexpr float EPSF = 1e-6f;

static constexpr int B_ROWS          = 65536;
static constexpr int DH              = 512;            // f32 per row
static constexpr int THREADS         = 256;            // 8 wave32s
static constexpr int WAVES           = THREADS / 32;
static constexpr int ROWS_PER_TILE   = 8;              // one row per wave
static constexpr int TILE_F32        = ROWS_PER_TILE * DH;     // 4096 floats, 16KB
static constexpr int NTILES          = B_ROWS / ROWS_PER_TILE; // 8192
static constexpr int NBLOCKS         = 1024;
static constexpr int TILES_PER_BLOCK = NTILES / NBLOCKS;       // 8

// Issue one TDM load of a contiguous 1-D tile (TILE_F32 f32) into LDS.
// Descriptor packing per CDNA5 ISA:
//  group0: [1:0]=count=1, [63:32]=lds_addr, [120:64]=global_addr, [127:126]=type(2)
//  group1: [17:16]=data_size(2 => 4B), [79:48]=tensor_dim0, [111:80]=tensor_dim1,
//          [127:112]=tile_dim0, [143:128]=tile_dim1, [207:160]=tensor_dim0_stride
__device__ __forceinline__ void tdm_load_tile(const float* gsrc, unsigned lds_byte_off) {
  const uint64_t ga = (uint64_t)(uintptr_t)gsrc;

  uint32x4 g0;
  g0[0] = 1u;                                               // count=1, user mode
  g0[1] = lds_byte_off;                                     // lds_addr (bytes)
  g0[2] = (unsigned)ga;                                     // global_addr[31:0]
  g0[3] = ((unsigned)(ga >> 32) & 0x01FFFFFFu) | (2u << 30);// addr[56:32] | type=2

  int32x8 g1;
  g1[0] = (int)(2u << 16);                 // data_size=2 (4 bytes); mask/flags=0
  g1[1] = (int)(((unsigned)TILE_F32 & 0xFFFFu) << 16);      // tensor_dim0 lo16
  g1[2] = (int)((((unsigned)TILE_F32 >> 16) & 0xFFFFu)      // tensor_dim0 hi16
                | (1u << 16));                              // tensor_dim1 lo16 = 1
  g1[3] = (int)(((unsigned)TILE_F32 & 0xFFFFu) << 16);      // tile_dim0 = TILE_F32
  g1[4] = 1;                                                // tile_dim1 = 1
  g1[5] = TILE_F32;                                         // tensor_dim0_stride lo32
  g1[6] = 0;
  g1[7] = 0;

  int32x4 gz = {0, 0, 0, 0};

#if defined(__clang_major__) && (__clang_major__ >= 23)
  int32x8 gz8 = {0, 0, 0, 0, 0, 0, 0, 0};
  __builtin_amdgcn_tensor_load_to_lds(g0, g1, gz, gz, gz8, 0);
#else
  __builtin_amdgcn_tensor_load_to_lds(g0, g1, gz, gz, 0);
#endif
}

__global__ void __launch_bounds__(THREADS)
siam_partial(const float* __restrict__ x1, const float* __restrict__ x2,
             const float* __restrict__ y1, const float* __restrict__ y2,
             const float* __restrict__ yp1, float* __restrict__ partial)
{
  __shared__ float4 sA[2][ROWS_PER_TILE][DH / 4];   // 32 KB
  __shared__ float4 sB[2][ROWS_PER_TILE][DH / 4];   // 32 KB
  __shared__ float  wacc[WAVES];

  const int tid   = threadIdx.x;
  const int wid   = tid >> 5;
  const int lane  = tid & 31;
  const int tile0 = blockIdx.x * TILES_PER_BLOCK;

  // Prologue: wave 0 kicks off tile 0 into buffer 0 (TDM ignores EXEC; one
  // issue per wave => guard to a single wave).
  if (tid < 32) {
    tdm_load_tile(x1 + (size_t)tile0 * TILE_F32,
                  (unsigned)(uintptr_t)&sA[0][0][0]);
    tdm_load_tile(x2 + (size_t)tile0 * TILE_F32,
                  (unsigned)(uintptr_t)&sB[0][0][0]);
  }

  float acc = 0.0f;

  for (int i = 0; i < TILES_PER_BLOCK; ++i) {
    const int buf = i & 1;

    if (tid < 32) {
      if (i + 1 < TILES_PER_BLOCK) {
        const int t  = tile0 + i + 1;
        const int nb = buf ^ 1;
        tdm_load_tile(x1 + (size_t)t * TILE_F32,
                      (unsigned)(uintptr_t)&sA[nb][0][0]);
        tdm_load_tile(x2 + (size_t)t * TILE_F32,
                      (unsigned)(uintptr_t)&sB[nb][0][0]);
        // 2 newest ops may remain outstanding; tensor ops complete in order,
        // so TENSORcnt<=2 means tile i has fully landed in LDS.
        __builtin_amdgcn_s_wait_tensorcnt(2);
      } else {
        __builtin_amdgcn_s_wait_tensorcnt(0);
      }
    }
    __syncthreads();   // publish tile i to all waves

    // Wave `wid` reduces row `wid` of this tile out of LDS.
    float sum = 0.0f;
    int   eq  = 1;
#pragma unroll
    for (int j = 0; j < (DH / 4) / 32; ++j) {
      const float4 a = sA[buf][wid][j * 32 + lane];
      const float4 b = sB[buf][wid][j * 32 + lane];
      const float d0 = a.x - b.x, d1 = a.y - b.y;
      const float d2 = a.z - b.z, d3 = a.w - b.w;
      sum += d0 * d0 + d1 * d1 + d2 * d2 + d3 * d3;
      eq  &= (int)((a.x == b.x) & (a.y == b.y) & (a.z == b.z) & (a.w == b.w));
    }
    // wave32 tree reduction
#pragma unroll
    for (int off = 16; off > 0; off >>= 1) {
      sum += __shfl_down(sum, off, 32);
      eq  &= __shfl_down(eq,  off, 32);
    }

    if (lane == 0) {
      const int row = (tile0 + i) * ROWS_PER_TILE + wid;
      const float dx = sqrtf(sum + EPSF);
      const float a0 = y1[2 * row + 0], a1 = y1[2 * row + 1];
      const float e0 = a0 - y2[2 * row + 0], e1 = a1 - y2[2 * row + 1];
      const float dy = sqrtf(e0 * e0 + e1 * e1 + EPSF);
      const float w  = eq ? 1.0f : dx;
      const float g0 = a0 - yp1[2 * row + 0], g1 = a1 - yp1[2 * row + 1];
      const float t  = dx - dy;
      acc += t * t / w + g0 * g0 + g1 * g1;
    }
    __syncthreads();   // everyone done reading `buf` before it is refilled
  }

  if (lane == 0) wacc[wid] = acc;
  __syncthreads();
  if (tid == 0) {
    float s = 0.0f;
#pragma unroll
    for (int w = 0; w < WAVES; ++w) s += wacc[w];
    partial[blockIdx.x] = s;
  }
}

__global__ void __launch_bounds__(THREADS)
siam_final(const float* __restrict__ partial, float* __restrict__ out)
{
  __shared__ float wacc[WAVES];
  const int tid = threadIdx.x, wid = tid >> 5, lane = tid & 31;

  float s = 0.0f;
  for (int i = tid; i < NBLOCKS; i += THREADS) s += partial[i];
#pragma unroll
  for (int off = 16; off > 0; off >>= 1) s += __shfl_down(s, off, 32);
  if (lane == 0) wacc[wid] = s;
  __syncthreads();
  if (tid == 0) {
    float t = 0.0f;
#pragma unroll
    for (int w = 0; w < WAVES; ++w) t += wacc[w];
    out[0] = t;   // full overwrite: deterministic, poison-safe
  }
}

extern "C" void kernel_launch(void* const* d_in, const int* in_sizes, int n_in,
                              void* d_out, int out_size, void* d_ws, size_t ws_size,
                              hipStream_t stream) {
  (void)in_sizes; (void)n_in; (void)out_size; (void)ws_size;
  const float* x1  = (const float*)d_in[0];
  const float* x2  = (const float*)d_in[1];
  const float* y1  = (const float*)d_in[2];
  const float* y2  = (const float*)d_in[3];
  const float* yp1 = (const float*)d_in[4];
  float* out       = (float*)d_out;
  float* ws        = (float*)d_ws;   // NBLOCKS floats of scratch

  siam_partial<<<NBLOCKS, THREADS, 0, stream>>>(x1, x2, y1, y2, yp1, ws);
  siam_final<<<1, THREADS, 0, stream>>>(ws, out);
}